// LearnableTMaxAvgPool2d_68685116997902
// MI455X (gfx1250) — compile-verified
//
#include <hip/hip_runtime.h>
#include <math.h>

// LearnableTMaxAvgPool2d: 2x2/stride-2 pooling, out = (max >= sigmoid(T[c])) ? max : mean(top2)
// x: (32,128,112,112) f32, T_raw: (128,) f32, out: (32,128,56,56) f32.
// Pure streaming op (~257 MB traffic) -> HBM-bound; optimize for b128 NT loads/stores.

typedef float f4 __attribute__((ext_vector_type(4)));

#define N_  32
#define C_  128
#define H_  112
#define W_  112
#define HO_ 56
#define WO_ 56
#define CHUNKS_ (WO_ / 4)   // 14 float4 chunks per output row

__global__ __launch_bounds__(256) void
LearnableTMaxAvgPool2d_kernel(const float* __restrict__ x,
                              const float* __restrict__ T_raw,
                              float* __restrict__ out)
{
    const long total = (long)N_ * C_ * HO_ * CHUNKS_;   // 3,211,264 work items
    long tid = (long)blockIdx.x * blockDim.x + threadIdx.x;
    if (tid >= total) return;

    int  chunk = (int)(tid % CHUNKS_);       // which float4 of the output row
    long row   = tid / CHUNKS_;              // flat (n, c, ho)
    int  c     = (int)((row / HO_) % C_);    // channel for threshold

    // Input: two consecutive rows of the 112-wide image; 8 floats per row per thread.
    const float* in = x + row * (2L * W_) + (long)chunk * 8;

    // All four loads are 16B-aligned (row stride 448B, chunk step 32B).
    f4 t0 = __builtin_nontemporal_load((const f4*)(in));
    f4 t1 = __builtin_nontemporal_load((const f4*)(in + 4));
    f4 b0 = __builtin_nontemporal_load((const f4*)(in + W_));
    f4 b1 = __builtin_nontemporal_load((const f4*)(in + W_ + 4));

    // Streaming prefetch ~8 row-pairs (7 KB) ahead: gfx1250 global_prefetch_b8.
    __builtin_prefetch(in + 8L * 2 * W_, 0, 1);

    // Per-channel threshold: sigmoid(T_raw[c]).  2 transcendentals per 80B of
    // memory traffic -> negligible vs. the bandwidth roofline.
    float traw = T_raw[c];
    float T    = 1.0f / (1.0f + __expf(-traw));

    float tops[8] = { t0.x, t0.y, t0.z, t0.w, t1.x, t1.y, t1.z, t1.w };
    float bots[8] = { b0.x, b0.y, b0.z, b0.w, b1.x, b1.y, b1.z, b1.w };

    f4 res;
#pragma unroll
    for (int j = 0; j < 4; ++j) {
        float a = tops[2 * j], b = tops[2 * j + 1];
        float e = bots[2 * j], d = bots[2 * j + 1];
        // Branchless top-2 of 4 via sort network.
        float hi1 = fmaxf(a, b), lo1 = fminf(a, b);
        float hi2 = fmaxf(e, d), lo2 = fminf(e, d);
        float m1  = fmaxf(hi1, hi2);                         // max of window
        float losr = (hi1 >= hi2) ? lo1 : lo2;               // lo of winning pair
        float m2  = fmaxf(fminf(hi1, hi2), losr);            // 2nd largest
        float avg = 0.5f * (m1 + m2);                        // mean of top-2
        res[j] = (m1 >= T) ? m1 : avg;
    }

    // 16B-aligned NT store of 4 pooled outputs.
    __builtin_nontemporal_store(res, (f4*)(out + row * WO_ + (long)chunk * 4));
}

extern "C" void kernel_launch(void* const* d_in, const int* in_sizes, int n_in,
                              void* d_out, int out_size, void* d_ws, size_t ws_size,
                              hipStream_t stream)
{
    (void)in_sizes; (void)n_in; (void)out_size; (void)d_ws; (void)ws_size;

    const float* x     = (const float*)d_in[0];   // (32,128,112,112) f32
    const float* T_raw = (const float*)d_in[1];   // (128,) f32
    float*       out   = (float*)d_out;           // (32,128,56,56) f32

    const long total  = (long)N_ * C_ * HO_ * CHUNKS_;       // 3,211,264
    const int  block  = 256;                                  // 8 wave32 waves
    const int  grid   = (int)((total + block - 1) / block);   // 12,544 blocks

    LearnableTMaxAvgPool2d_kernel<<<grid, block, 0, stream>>>(x, T_raw, out);
}